// MorphologicalLayer_43525198578010
// MI455X (gfx1250) — compile-verified
//
#include <hip/hip_runtime.h>
#include <hip/hip_bf16.h>

// Morphological dilation (max-plus 3x3, zero pad) for x=(4,64,512,512) f32,
// per-channel weights (64,3,3). Pure HBM-bandwidth-bound (~536MB @ 23.3TB/s).
// CDNA5 path: async global->LDS B128 staging (ASYNCcnt) + wave32 tiling.
// TH=16 keeps halo read amplification at (TH+2)/TH = 1.125x; sliding-window
// registers cut LDS reads 3x vs re-reading all 3 stencil rows.

#define HH 512
#define WW 512
#define TH 16                   // output rows per block
#define LDSW 516                // floats per LDS row: 516*4=2064B (16B aligned, bank-rotating)
#define NROWS (TH + 2)          // staged input rows incl. vertical halo (even)

typedef __attribute__((address_space(3))) float lds_f32_t;

// Load the 4 stencil columns (c0-1 .. c0+2) of one staged row, zero at edges.
__device__ __forceinline__ void ldrow(const float* __restrict__ lr, int c0, float v[4]) {
    v[0] = (c0 > 0)      ? lr[c0 - 1] : 0.0f;   // left zero-pad
    v[1] = lr[c0];
    v[2] = lr[c0 + 1];
    v[3] = (c0 + 2 < WW) ? lr[c0 + 2] : 0.0f;   // right zero-pad
}

__global__ __launch_bounds__(256) void morph_dilate3x3(
    const float* __restrict__ x,
    const float* __restrict__ wts,
    float* __restrict__ out)
{
    __shared__ float tile[NROWS * LDSW];

    const int tid = threadIdx.x;
    const int p   = blockIdx.y;           // plane = b*64 + c (contiguous in memory)
    const int ch  = p & 63;
    const int r0  = blockIdx.x * TH;      // first output row of this block

    const float* plane = x + (size_t)p * (HH * WW);

    // LDS byte offset of the tile base (addrspace(3) pointer value == LDS offset)
    const unsigned ldsBase = (unsigned)(uintptr_t)(lds_f32_t*)tile;

    // ---- stage NROWS input rows into LDS ----
    // 256 threads: half-block per row-pair element, 128 lanes x 16B = one 2048B row.
    const int half   = tid >> 7;          // row within pair (wave-uniform)
    const int lane16 = tid & 127;         // 16-byte chunk within the row
#pragma unroll
    for (int pass = 0; pass < NROWS / 2; ++pass) {
        const int rr = pass * 2 + half;
        const int gr = r0 - 1 + rr;       // global input row (with top halo)
        const unsigned loff = (unsigned)(rr * LDSW + lane16 * 4) * 4u;
        if (gr >= 0 && gr < HH) {
            const unsigned long long gaddr =
                (unsigned long long)(uintptr_t)(plane + (size_t)gr * WW + lane16 * 4);
            const unsigned laddr = ldsBase + loff;
            // CDNA5 async DMA: copies 16 bytes from the global address in gaddr to the LDS address in laddr, tracked on ASYNCcnt
            asm volatile("global_load_async_to_lds_b128 %0, %1, off"
                         :: "v"(laddr), "v"(gaddr)
                         : "memory");
        } else {
            // zero-pad halo rows (nn.Unfold pads with zeros; 0 + w still
            // participates in the max, so these must be real zeros)
            *(float4*)((char*)tile + loff) = make_float4(0.f, 0.f, 0.f, 0.f);
        }
    }

    // make this wave's async LDS writes complete, then block-wide visibility
    asm volatile("s_wait_asynccnt 0x0" ::: "memory");
    __syncthreads();

    // ---- per-channel 3x3 weights (uniform across block) ----
    float w[9];
    const float* wp = wts + ch * 9;
#pragma unroll
    for (int i = 0; i < 9; ++i) w[i] = wp[i];

    // ---- compute: 2 output pixels per thread per row, vertical sliding window ----
    const int c0 = tid * 2;               // 0..510, even -> 8B-aligned stores
    float* oplane = out + (size_t)p * (HH * WW);

    float ra[4], rb[4], rc[4];
    ldrow(&tile[0 * LDSW], c0, ra);
    ldrow(&tile[1 * LDSW], c0, rb);

#pragma unroll
    for (int orow = 0; orow < TH; ++orow) {
        ldrow(&tile[(orow + 2) * LDSW], c0, rc);

        float m0 = ra[0] + w[0];
        float m1 = ra[1] + w[0];
        m0 = fmaxf(m0, ra[1] + w[1]);  m1 = fmaxf(m1, ra[2] + w[1]);
        m0 = fmaxf(m0, ra[2] + w[2]);  m1 = fmaxf(m1, ra[3] + w[2]);
        m0 = fmaxf(m0, rb[0] + w[3]);  m1 = fmaxf(m1, rb[1] + w[3]);
        m0 = fmaxf(m0, rb[1] + w[4]);  m1 = fmaxf(m1, rb[2] + w[4]);
        m0 = fmaxf(m0, rb[2] + w[5]);  m1 = fmaxf(m1, rb[3] + w[5]);
        m0 = fmaxf(m0, rc[0] + w[6]);  m1 = fmaxf(m1, rc[1] + w[6]);
        m0 = fmaxf(m0, rc[1] + w[7]);  m1 = fmaxf(m1, rc[2] + w[7]);
        m0 = fmaxf(m0, rc[2] + w[8]);  m1 = fmaxf(m1, rc[3] + w[8]);

        *(float2*)(oplane + (size_t)(r0 + orow) * WW + c0) = make_float2(m0, m1);

        // rotate the window (full unroll -> pure SSA renaming, no v_movs)
#pragma unroll
        for (int t = 0; t < 4; ++t) { ra[t] = rb[t]; rb[t] = rc[t]; }
    }
}

extern "C" void kernel_launch(void* const* d_in, const int* in_sizes, int n_in,
                              void* d_out, int out_size, void* d_ws, size_t ws_size,
                              hipStream_t stream) {
    (void)in_sizes; (void)n_in; (void)out_size; (void)d_ws; (void)ws_size;
    const float* x   = (const float*)d_in[0];   // (4,64,512,512) f32
    const float* wts = (const float*)d_in[1];   // (64,3,3) f32
    float*       out = (float*)d_out;           // (4,64,512*512) f32

    dim3 grid(HH / TH, 4 * 64);                 // 32 row-blocks x 256 planes
    morph_dilate3x3<<<grid, dim3(256), 0, stream>>>(x, wts, out);
}